// LoRAEmbedding_40355512714073
// MI455X (gfx1250) — compile-verified
//
#include <hip/hip_runtime.h>

typedef __attribute__((ext_vector_type(2))) float v2f;
typedef __attribute__((ext_vector_type(4))) float v4f;
typedef __attribute__((ext_vector_type(8))) float v8f;

#define LORA_D 1024
#define LORA_R 8
#define LORA_SCALING 2.0f   // 16.0 / 8.0
#define WAVES_PER_BLOCK 4
#define BLOCK_THREADS (WAVES_PER_BLOCK * 32)
#define DTILES_PER_WAVE (LORA_D / (WAVES_PER_BLOCK * 16))   // 16

// One block = one 16-token tile. Each wave covers 256 columns of D.
// delta(16x16) computed with V_WMMA_F32_16X16X4_F32 (two k-steps for R=8),
// bounced through LDS so the base-add + store phase is float4-coalesced.
__global__ __launch_bounds__(BLOCK_THREADS)
void lora_embed_wmma(const int* __restrict__ x,
                     const float* __restrict__ W,
                     const float* __restrict__ A,    // (R, V)
                     const float* __restrict__ Bm,   // (D, R)
                     float* __restrict__ out,        // (BS, D)
                     int V)
{
    __shared__ float sB[LORA_D * LORA_R];               // 32 KB: lora_B staged once
    __shared__ int   sX[16];                            // token ids of this tile
    __shared__ float sD[WAVES_PER_BLOCK * 16 * 16];     // per-wave delta tile (1 KB each)

    const int tbase = blockIdx.x * 16;

    for (int i = threadIdx.x; i < LORA_D * LORA_R; i += BLOCK_THREADS)
        sB[i] = Bm[i];
    if (threadIdx.x < 16)
        sX[threadIdx.x] = x[tbase + threadIdx.x];
    __syncthreads();

    const int lane  = threadIdx.x & 31;
    const int wave  = threadIdx.x >> 5;
    const int n     = lane & 15;            // row (for A) / column (for B,C) selector
    const int khalf = (lane >> 4) << 1;     // 0 or 2: K pair within a 16x4 block

    // ---- A fragments: 16x8 (tokens x R) as two 16x4 K-blocks ----
    // ISA 32-bit A 16x4 layout: vgpr0 = K=khalf, vgpr1 = K=khalf+1, M = lane&15
    const int tokm = sX[n];
    v2f a0, a1;
    a0.x = A[(size_t)(khalf + 0) * V + tokm];
    a0.y = A[(size_t)(khalf + 1) * V + tokm];
    a1.x = A[(size_t)(khalf + 4) * V + tokm];
    a1.y = A[(size_t)(khalf + 5) * V + tokm];

    // ---- phase-2 mapping: lane owns two contiguous float4 chunks of one row ----
    const int c0     = lane * 2;            // first 4-float chunk index (0..63)
    const int row2   = c0 >> 2;             // tile row handled in phase 2 (0..15)
    const int col2   = (c0 & 3) * 4;        // starting column of first chunk
    const int trow2  = sX[row2];            // W row for phase 2
    const int rowoff = (lane >> 4) << 3;    // C-fragment: rows j+0 or j+8

    float* myD = &sD[wave * 256];

    // Cap unroll: 4 C tiles in flight keeps VGPR pressure low enough for
    // multiple resident waves (HBM latency hiding) while still forming
    // good ds/wmma clauses.
    #pragma unroll 4
    for (int dt = 0; dt < DTILES_PER_WAVE; ++dt) {
        const int dbase = (wave * DTILES_PER_WAVE + dt) * 16;

        // ---- B fragment from LDS: B[k][n] = lora_B[(dbase+n), k] ----
        const float* bp = &sB[(dbase + n) * LORA_R + khalf];
        v2f b0, b1;
        b0.x = bp[0]; b0.y = bp[1];
        b1.x = bp[4]; b1.y = bp[5];

        v8f c = {};
        c = __builtin_amdgcn_wmma_f32_16x16x4_f32(false, a0, false, b0,
                                                  (short)0, c, false, false);
        c = __builtin_amdgcn_wmma_f32_16x16x4_f32(false, a1, false, b1,
                                                  (short)0, c, false, false);

        // ---- scatter delta tile to per-wave LDS scratch (C-fragment layout) ----
        #pragma unroll
        for (int j = 0; j < 8; ++j)
            myD[(j + rowoff) * 16 + n] = c[j];
        // same-wave LDS ops are in-order; compiler inserts the needed s_wait_dscnt

        // ---- coalesced base-add + store: two float4 chunks per lane ----
        const v4f d0 = *(const v4f*)&myD[row2 * 16 + col2];
        const v4f d1 = *(const v4f*)&myD[row2 * 16 + col2 + 4];
        const float* wrow = W + (size_t)trow2 * LORA_D + dbase;
        const v4f w0 = *(const v4f*)(wrow + col2);
        const v4f w1 = *(const v4f*)(wrow + col2 + 4);
        v4f o0, o1;
        #pragma unroll
        for (int q = 0; q < 4; ++q) {
            o0[q] = w0[q] + LORA_SCALING * d0[q];
            o1[q] = w1[q] + LORA_SCALING * d1[q];
        }
        float* orow = out + (size_t)(tbase + row2) * LORA_D + dbase;
        *(v4f*)(orow + col2)     = o0;
        *(v4f*)(orow + col2 + 4) = o1;
    }
}

extern "C" void kernel_launch(void* const* d_in, const int* in_sizes, int n_in,
                              void* d_out, int out_size, void* d_ws, size_t ws_size,
                              hipStream_t stream) {
    const int*   x  = (const int*)  d_in[0];   // (B, S) int32
    const float* W  = (const float*)d_in[1];   // (V, D) f32
    const float* A  = (const float*)d_in[2];   // (R, V) f32
    const float* Bm = (const float*)d_in[3];   // (D, R) f32
    float* out = (float*)d_out;                // (B, S, D) f32

    const int BS = in_sizes[0];                // 16384 tokens
    const int V  = in_sizes[2] / LORA_R;       // 50257

    const int tiles = BS / 16;                 // 1024 token tiles
    lora_embed_wmma<<<tiles, BLOCK_THREADS, 0, stream>>>(x, W, A, Bm, out, V);
}